// LFReparam_31808527794661
// MI455X (gfx1250) — compile-verified
//
#include <hip/hip_runtime.h>
#include <cstdint>

#define XRES 512
#define YRES 512
#define UV 9
#define RADIUS 4
#define HH (YRES * UV) /* 4608 */
#define WW (XRES * UV) /* 4608 */
#define CH 3

#define TPB 288                 /* 9 wave32 per block */
#define ROW_BYTES (WW * 4)      /* 18432 bytes per staged row */

__global__ __launch_bounds__(TPB) void lf_reparam_kernel(
    const float* __restrict__ x, const float* __restrict__ alphaPtr,
    float* __restrict__ out)
{
    // Two full input rows staged in LDS (36 KB).
    __shared__ __align__(16) float smem[2 * WW];

    const int tid = threadIdx.x;
    const int blk = blockIdx.x;        // blk = c*HH + h
    const int c   = blk / HH;
    const int h   = blk - c * HH;
    const int y   = h / UV;
    const int v   = h - y * UV;

    const float alpha  = alphaPtr[0];
    const float nalpha = -alpha;

    // Vertical taps/weights: uniform per block (depend only on v, y) -> SALU.
    const float fy  = nalpha * (float)(v - RADIUS);
    const int   iy  = (int)floorf(fy);
    const float wy2 = fy - (float)iy;
    const float wy1 = 1.0f - wy2;
    int y1 = y + iy;     y1 = min(max(y1, 0), YRES - 1);
    int y2 = y + iy + 1; y2 = min(max(y2, 0), YRES - 1);
    const int r1 = y1 * UV + v;
    const int r2 = y2 * UV + v;

    const size_t plane = (size_t)c * (size_t)HH * (size_t)WW;
    const float* src1 = x + plane + (size_t)r1 * WW;
    const float* src2 = x + plane + (size_t)r2 * WW;

    // LDS byte offset of smem base (low 32 bits of the generic address).
    const unsigned sbase = (unsigned)(uintptr_t)(&smem[0]);

    // Async-stage row r1 -> smem[0:WW), row r2 -> smem[WW:2WW).
    // 1152 float4 chunks per row, 288 threads -> 4 chunks/thread/row.
#pragma unroll
    for (int k = 0; k < 4; ++k) {
        const int ci = tid + k * TPB;                       // 0..1151
        const unsigned laddr = sbase + (unsigned)(ci * 16);
        const uint64_t ga = (uint64_t)(uintptr_t)(src1 + ci * 4);
        asm volatile("global_load_async_to_lds_b128 %0, %1, off"
                     :: "v"(laddr), "v"(ga) : "memory");
    }
#pragma unroll
    for (int k = 0; k < 4; ++k) {
        const int ci = tid + k * TPB;
        const unsigned laddr = sbase + (unsigned)(ROW_BYTES + ci * 16);
        const uint64_t ga = (uint64_t)(uintptr_t)(src2 + ci * 4);
        asm volatile("global_load_async_to_lds_b128 %0, %1, off"
                     :: "v"(laddr), "v"(ga) : "memory");
    }
    asm volatile("s_wait_asynccnt 0" ::: "memory");
    __syncthreads();

    const float* __restrict__ row0 = smem;
    const float* __restrict__ row1 = smem + WW;
    float* __restrict__ dst = out + plane + (size_t)h * WW;

    // Per-k stride is TPB*4 = 1152 = 128*9: an exact multiple of UV, so for a
    // fixed thread the lenslet coordinate u (and the horizontal tap offsets /
    // weights) are the SAME for all k. Hoist them out of the k loop; only the
    // macro-pixel coordinate advances by 128 per k.
    int   u_j[4];     // lenslet u per j lane
    int   xb_j[4];    // base macro-pixel x (k = 0)
    int   ix_j[4];    // floor(-alpha*(u-4))
    float wx1_j[4], wx2_j[4];
#pragma unroll
    for (int j = 0; j < 4; ++j) {
        const int w0 = tid * 4 + j;       // k = 0 output column
        const int xm = w0 / UV;
        const int u  = w0 - xm * UV;
        const float fx  = nalpha * (float)(u - RADIUS);
        const int   ix  = (int)floorf(fx);
        u_j[j]   = u;
        xb_j[j]  = xm;
        ix_j[j]  = ix;
        wx2_j[j] = fx - (float)ix;
        wx1_j[j] = 1.0f - wx2_j[j];
    }

    // 4 float4 outputs per thread (16 scalars); coalesced b128 stores.
#pragma unroll
    for (int k = 0; k < 4; ++k) {
        const int w4 = tid * 4 + k * (TPB * 4);
        float tmp[4];
#pragma unroll
        for (int j = 0; j < 4; ++j) {
            const int xm = xb_j[j] + k * 128;
            int x1 = xm + ix_j[j];     x1 = min(max(x1, 0), XRES - 1);
            int x2 = xm + ix_j[j] + 1; x2 = min(max(x2, 0), XRES - 1);
            const int col1 = u_j[j] + x1 * UV;
            const int col2 = u_j[j] + x2 * UV;
            const float a0 = row0[col1], b0 = row0[col2];
            const float a1 = row1[col1], b1 = row1[col2];
            tmp[j] = wy1 * (wx1_j[j] * a0 + wx2_j[j] * b0)
                   + wy2 * (wx1_j[j] * a1 + wx2_j[j] * b1);
        }
        *(float4*)(dst + w4) = make_float4(tmp[0], tmp[1], tmp[2], tmp[3]);
    }
}

extern "C" void kernel_launch(void* const* d_in, const int* in_sizes, int n_in,
                              void* d_out, int out_size, void* d_ws, size_t ws_size,
                              hipStream_t stream) {
    const float* x        = (const float*)d_in[0];
    const float* alphaPtr = (const float*)d_in[1];
    float* out            = (float*)d_out;

    dim3 grid(CH * HH);   // 13824 blocks: one per (channel, output row)
    dim3 block(TPB);      // 288 threads = 9 wave32
    hipLaunchKernelGGL(lf_reparam_kernel, grid, block, 0, stream,
                       x, alphaPtr, out);
}